// GrayscaleBlock_9766755631740
// MI455X (gfx1250) — compile-verified
//
#include <hip/hip_runtime.h>

typedef __attribute__((ext_vector_type(2))) float v2f;
typedef __attribute__((ext_vector_type(8))) float v8f;

#define BATCH    4
#define CHANNELS 256
#define LATENT   512
#define DIM      32
#define SPATIAL  (DIM * DIM * DIM)   // 32768
#define ODIM     64

// ---------------------------------------------------------------------------
// Kernel 1: style = istyle @ style_w^T + style_b ; mod = conv_w * (style + 1)
// One block per batch, one thread per channel. istyle row staged in LDS.
// ---------------------------------------------------------------------------
__global__ __launch_bounds__(CHANNELS)
void style_mod_kernel(const float* __restrict__ istyle,
                      const float* __restrict__ style_w,
                      const float* __restrict__ style_b,
                      const float* __restrict__ conv_w,
                      float* __restrict__ mod) {
    __shared__ float s_is[LATENT];
    const int b = blockIdx.x;
    const int c = threadIdx.x;
    for (int i = threadIdx.x; i < LATENT; i += CHANNELS)
        s_is[i] = istyle[b * LATENT + i];
    __syncthreads();

    const float4* wrow = (const float4*)(style_w + c * LATENT);
    float acc = style_b[c];
    #pragma unroll 4
    for (int l = 0; l < LATENT / 4; ++l) {
        float4 w4 = wrow[l];
        acc += s_is[4 * l + 0] * w4.x + s_is[4 * l + 1] * w4.y +
               s_is[4 * l + 2] * w4.z + s_is[4 * l + 3] * w4.w;
    }
    mod[b * CHANNELS + c] = conv_w[c] * (acc + 1.0f);
}

// ---------------------------------------------------------------------------
// Kernel 2: y0[b,s] = sum_c x[b,c,s] * mod[b,c] + prev_rgb[b,s]
// One wave handles a 16-spatial tile; channel loop in steps of 4 via
// V_WMMA_F32_16X16X4_F32 (A = 16 spatial x 4 chan, B = mod broadcast to all
// 16 columns). D column 0 extracted from lanes 0 (M=0..7) and 16 (M=8..15).
// Memory-bound: streams all 134 MB of x once.
// ---------------------------------------------------------------------------
__global__ __launch_bounds__(256)
void einsum_wmma_kernel(const float* __restrict__ x,
                        const float* __restrict__ prev,
                        const float* __restrict__ mod,
                        float* __restrict__ y0) {
    const int lane = threadIdx.x & 31;
    const int wave = threadIdx.x >> 5;
    const int tile = blockIdx.x * 8 + wave;          // 8192 tiles total
    const int tilesPerBatch = SPATIAL / 16;          // 2048
    const int b  = tile / tilesPerBatch;
    const int s0 = (tile % tilesPerBatch) * 16;

    const int half = lane >> 4;    // 0: lanes 0-15 (K=0,1) ; 1: lanes 16-31 (K=2,3)
    const int m    = lane & 15;    // row M (for A) / column N (for B)

    const float* xb   = x   + (size_t)b * CHANNELS * SPATIAL;
    const float* modb = mod + b * CHANNELS;

    v8f acc = {};
    for (int c0 = 0; c0 < CHANNELS; c0 += 4) {
        const int cA = c0 + 2 * half;
        v2f a, bb;
        a.x  = xb[(size_t)cA       * SPATIAL + s0 + m];   // A[m, cA]
        a.y  = xb[(size_t)(cA + 1) * SPATIAL + s0 + m];   // A[m, cA+1]
        bb.x = modb[cA];                                  // B[cA, n] (all n equal)
        bb.y = modb[cA + 1];
        acc = __builtin_amdgcn_wmma_f32_16x16x4_f32(
            /*neg_a=*/false, a, /*neg_b=*/false, bb,
            /*c_mod=*/(short)0, acc, /*reuse_a=*/false, /*reuse_b=*/false);
    }

    // Column N=0 of D: lane 0 holds M=0..7 in acc[0..7], lane 16 holds M=8..15.
    if (m == 0) {
        const int base = s0 + 8 * half;
        const float4* pv = (const float4*)(prev + (size_t)b * SPATIAL + base);
        float4 p0 = pv[0], p1 = pv[1];
        float4 r0 = { acc[0] + p0.x, acc[1] + p0.y, acc[2] + p0.z, acc[3] + p0.w };
        float4 r1 = { acc[4] + p1.x, acc[5] + p1.y, acc[6] + p1.z, acc[7] + p1.w };
        float4* ov = (float4*)(y0 + (size_t)b * SPATIAL + base);
        ov[0] = r0;
        ov[1] = r1;
    }
}

// ---------------------------------------------------------------------------
// Kernel 3: fused trilinear x2 upsample (half-pixel) + [1,2,1]^3 blur.
// Composition is a separable 3-tap filter on y0 with parity-dependent
// weights: even p -> (0.3125, 0.625, 0.0625), odd p -> mirrored; indices
// edge-clamped (verified to match resize+replicate-pad boundary math).
// ---------------------------------------------------------------------------
__device__ __forceinline__ void axis_taps(int p, int* i3, float* w3) {
    const int m   = p >> 1;
    const int par = p & 1;
    i3[0] = (m - 1 < 0) ? 0 : m - 1;
    i3[1] = m;
    i3[2] = (m + 1 > DIM - 1) ? DIM - 1 : m + 1;
    w3[0] = par ? 0.0625f : 0.3125f;
    w3[1] = 0.625f;
    w3[2] = par ? 0.3125f : 0.0625f;
}

__global__ __launch_bounds__(256)
void upblur_kernel(const float* __restrict__ y0, float* __restrict__ out) {
    const int idx = blockIdx.x * 256 + threadIdx.x;   // 4*64^3 threads
    const int w = idx & 63;
    const int h = (idx >> 6) & 63;
    const int d = (idx >> 12) & 63;
    const int b = idx >> 18;

    const float* yb = y0 + (size_t)b * SPATIAL;

    int id[3], ih[3], iw[3];
    float wd[3], wh[3], ww[3];
    axis_taps(d, id, wd);
    axis_taps(h, ih, wh);
    axis_taps(w, iw, ww);

    float sum = 0.0f;
    #pragma unroll
    for (int i = 0; i < 3; ++i) {
        #pragma unroll
        for (int j = 0; j < 3; ++j) {
            const float* row = yb + (id[i] * DIM + ih[j]) * DIM;
            const float rs = ww[0] * row[iw[0]] + ww[1] * row[iw[1]] + ww[2] * row[iw[2]];
            sum += wd[i] * wh[j] * rs;
        }
    }
    out[idx] = sum;
}

// ---------------------------------------------------------------------------
extern "C" void kernel_launch(void* const* d_in, const int* in_sizes, int n_in,
                              void* d_out, int out_size, void* d_ws, size_t ws_size,
                              hipStream_t stream) {
    const float* x       = (const float*)d_in[0];  // (4,256,32,32,32)
    const float* prev    = (const float*)d_in[1];  // (4,1,32,32,32)
    const float* istyle  = (const float*)d_in[2];  // (4,512)
    const float* style_w = (const float*)d_in[3];  // (256,512)
    const float* style_b = (const float*)d_in[4];  // (256,)
    const float* conv_w  = (const float*)d_in[5];  // (1,256,1,1,1)
    float* out = (float*)d_out;                    // (4,1,64,64,64)

    float* mod = (float*)d_ws;                     // 1024 floats
    float* y0  = mod + BATCH * CHANNELS;           // 131072 floats

    // 1) per-(batch,channel) modulation weights
    style_mod_kernel<<<BATCH, CHANNELS, 0, stream>>>(istyle, style_w, style_b, conv_w, mod);

    // 2) WMMA channel reduction + prev_rgb add  (8192 waves, 8 waves/block)
    einsum_wmma_kernel<<<(BATCH * (SPATIAL / 16)) / 8, 256, 0, stream>>>(x, prev, mod, y0);

    // 3) fused upsample x2 + blur
    upblur_kernel<<<(BATCH * ODIM * ODIM * ODIM) / 256, 256, 0, stream>>>(y0, out);
}